// RoutingLayer_70308614635673
// MI455X (gfx1250) — compile-verified
//
#include <hip/hip_runtime.h>
#include <math.h>

typedef __attribute__((ext_vector_type(2))) float v2f;
typedef __attribute__((ext_vector_type(8))) float v8f;

#define SITES  2      // sites per block
#define NROWS  80     // 64 neighbor rows + 2 self rows + 14 zero-pad (5 M-tiles)
#define STRIDE 132    // padded row stride in floats (bank-conflict avoidance)
#define L2EPS  1e-12f

__global__ __launch_bounds__(256)
void routing_fused_kernel(const float* __restrict__ self_v,   // [8192,128]
                          const float* __restrict__ neigh_v,  // [8192,32,128]
                          const float* __restrict__ w1,       // [128,128] row-major (k,n)
                          const float* __restrict__ b1,       // [128]
                          const int*   __restrict__ max_iter_p,
                          float*       __restrict__ out)      // [8192,128]
{
    __shared__ float Xz[NROWS * STRIDE];        // X tile, later overwritten with Z
    __shared__ float bias_s[128];
    __shared__ float u_s[SITES][128];
    __shared__ float p_s[SITES][32 * 8];
    __shared__ float inv_s[SITES][8];

    const int tid   = threadIdx.x;
    const int site0 = blockIdx.x * SITES;

    // ---------------- Phase 0: stage inputs into LDS ----------------
    {
        const float4* ng = (const float4*)(neigh_v + (size_t)site0 * 32 * 128);
        for (int i = tid; i < (64 * 128) / 4; i += 256) {
            float4 v = ng[i];
            int flat = i * 4;
            int row = flat >> 7, col = flat & 127;
            float* d = &Xz[row * STRIDE + col];
            d[0] = v.x; d[1] = v.y; d[2] = v.z; d[3] = v.w;
        }
        if (tid < (SITES * 128) / 4) {
            const float4* sv = (const float4*)(self_v + (size_t)site0 * 128);
            float4 v = sv[tid];
            int flat = tid * 4;
            int row = 64 + (flat >> 7), col = flat & 127;
            float* d = &Xz[row * STRIDE + col];
            d[0] = v.x; d[1] = v.y; d[2] = v.z; d[3] = v.w;
        }
        for (int i = tid; i < (NROWS - 66) * 128; i += 256) {
            int row = 66 + (i >> 7), col = i & 127;
            Xz[row * STRIDE + col] = 0.0f;
        }
        if (tid < 128) bias_s[tid] = b1[tid];
    }
    __syncthreads();

    const int wave = tid >> 5;     // wave == N-stripe index nt
    const int lane = tid & 31;
    const int hi   = lane >> 4;    // 0: lanes 0-15, 1: lanes 16-31
    const int lo   = lane & 15;
    const int nt   = wave;

    // ---------------- Phase 1: GEMM  Z = X @ w1  (fp32 WMMA) ----------------
    // Preload this wave's B-fragments of w1 (column stripe nt) into registers.
    v2f bf[32];
    #pragma unroll
    for (int kk = 0; kk < 32; ++kk) {
        int k0 = kk * 4 + 2 * hi;
        bf[kk].x = w1[(k0 + 0) * 128 + nt * 16 + lo];
        bf[kk].y = w1[(k0 + 1) * 128 + nt * 16 + lo];
    }

    v8f zero = {0.f, 0.f, 0.f, 0.f, 0.f, 0.f, 0.f, 0.f};
    v8f acc[5];
    #pragma unroll
    for (int mt = 0; mt < 5; ++mt) acc[mt] = zero;

    for (int kk = 0; kk < 32; ++kk) {
        int k0 = kk * 4 + 2 * hi;
        #pragma unroll
        for (int mt = 0; mt < 5; ++mt) {
            v2f a = *(const v2f*)&Xz[(mt * 16 + lo) * STRIDE + k0];
            acc[mt] = __builtin_amdgcn_wmma_f32_16x16x4_f32(
                false, a, false, bf[kk], (short)0, acc[mt], false, false);
        }
    }
    __syncthreads();   // everyone done reading X before overwriting it with Z

    // ---------------- Phase 2: bias + relu, write Z over X tile ----------------
    {
        float bb = bias_s[nt * 16 + lo];
        #pragma unroll
        for (int mt = 0; mt < 5; ++mt) {
            #pragma unroll
            for (int r = 0; r < 8; ++r) {
                int m = mt * 16 + hi * 8 + r;          // C/D layout: VGPR r -> M = 8*hi + r
                float v = acc[mt][r] + bb;
                Xz[m * STRIDE + nt * 16 + lo] = v > 0.0f ? v : 0.0f;
            }
        }
    }
    __syncthreads();

    // ---------------- Phase 3: per-capsule L2 normalize rows 0..65 ----------------
    for (int cap = tid; cap < 66 * 8; cap += 256) {
        int row = cap >> 3, k = cap & 7;
        float* p = &Xz[row * STRIDE + k * 16];
        float ss = 0.f;
        #pragma unroll
        for (int c = 0; c < 16; ++c) ss += p[c] * p[c];
        float nrm = sqrtf(ss);
        float inv = 1.0f / (nrm > L2EPS ? nrm : L2EPS);
        #pragma unroll
        for (int c = 0; c < 16; ++c) p[c] *= inv;
    }
    __syncthreads();

    // ---------------- Phase 4: agreement routing, one wave per site ----------------
    if (wave < SITES) {
        const int s  = wave;
        const int nb = s * 32;                     // neighbor row base for this site
        const float* selfz = &Xz[(64 + s) * STRIDE];
        const int max_iter = *max_iter_p;
        float ucomp[4];                            // components lane + 32*j

        for (int it = 0; it < max_iter; ++it) {
            // p[n][k] with n = lane
            float p[8];
            if (it == 0) {
                #pragma unroll
                for (int k = 0; k < 8; ++k) p[k] = 0.125f;   // softmax(0) over K=8
            } else {
                float mx = -1e30f;
                #pragma unroll
                for (int k = 0; k < 8; ++k) {
                    const float* zr = &Xz[(nb + lane) * STRIDE + k * 16];
                    const float* ur = &u_s[s][k * 16];
                    float d = 0.f;
                    #pragma unroll
                    for (int c = 0; c < 16; ++c) d += zr[c] * ur[c];
                    p[k] = d;
                    mx = d > mx ? d : mx;
                }
                float sum = 0.f;
                #pragma unroll
                for (int k = 0; k < 8; ++k) { p[k] = __expf(p[k] - mx); sum += p[k]; }
                float r = 1.0f / sum;
                #pragma unroll
                for (int k = 0; k < 8; ++k) p[k] *= r;
            }
            #pragma unroll
            for (int k = 0; k < 8; ++k) p_s[s][lane * 8 + k] = p[k];

            // u[comp] = sum_n zn[n][comp] * p[n][comp/16] + self_zn[comp]
            #pragma unroll
            for (int j = 0; j < 4; ++j) {
                int comp = lane + 32 * j;
                int k = comp >> 4;
                float a = selfz[comp];
                for (int n = 0; n < 32; ++n)
                    a += Xz[(nb + n) * STRIDE + comp] * p_s[s][n * 8 + k];
                ucomp[j] = a;
                u_s[s][comp] = a;
            }
            if (it < max_iter - 1) {   // normalize u per capsule
                if (lane < 8) {
                    float ss = 0.f;
                    #pragma unroll
                    for (int c = 0; c < 16; ++c) {
                        float v = u_s[s][lane * 16 + c];
                        ss += v * v;
                    }
                    float nrm = sqrtf(ss);
                    inv_s[s][lane] = 1.0f / (nrm > L2EPS ? nrm : L2EPS);
                }
                #pragma unroll
                for (int j = 0; j < 4; ++j) {
                    int comp = lane + 32 * j;
                    float v = ucomp[j] * inv_s[s][comp >> 4];
                    u_s[s][comp] = v;
                }
            }
        }

        // ---------------- Phase 5: relu output ----------------
        float* o = out + (size_t)(site0 + s) * 128;
        #pragma unroll
        for (int j = 0; j < 4; ++j) {
            int comp = lane + 32 * j;
            float v = ucomp[j];
            o[comp] = v > 0.0f ? v : 0.0f;
        }
    }
}

extern "C" void kernel_launch(void* const* d_in, const int* in_sizes, int n_in,
                              void* d_out, int out_size, void* d_ws, size_t ws_size,
                              hipStream_t stream) {
    (void)in_sizes; (void)n_in; (void)out_size; (void)d_ws; (void)ws_size;
    const float* self_v  = (const float*)d_in[0];  // [128,64,128]
    const float* neigh_v = (const float*)d_in[1];  // [128,64,32,128]
    const float* w1      = (const float*)d_in[2];  // [128,128]
    const float* b1      = (const float*)d_in[3];  // [128]
    const int*   max_it  = (const int*)d_in[4];    // scalar
    float* out = (float*)d_out;                    // [128,64,128]

    const int n_sites = 128 * 64;                  // B*S = 8192
    dim3 grid(n_sites / SITES);                    // 4096 blocks
    dim3 block(256);                               // 8 wave32
    routing_fused_kernel<<<grid, block, 0, stream>>>(self_v, neigh_v, w1, b1, max_it, out);
}